// Attention_26379689132660
// MI455X (gfx1250) — compile-verified
//
#include <hip/hip_runtime.h>

typedef __attribute__((ext_vector_type(2))) float v2f;
typedef __attribute__((ext_vector_type(8))) float v8f;

#define TT   8      // time steps
#define CC   3      // channels in/out
#define DM   32     // d_model
#define HWN  65536  // H*W
#define POSW 16     // positions per wave/block

static __device__ __forceinline__ v8f wmma_f32k4(v2f a, v2f b) {
  v8f c = {};  // SRC2 = inline constant 0
  // (neg_a, A, neg_b, B, c_mod, C, reuse_a, reuse_b)
  return __builtin_amdgcn_wmma_f32_16x16x4_f32(false, a, false, b, (short)0, c, false, false);
}

// ---------------------------------------------------------------------------
// Kernel 1: compose the x-linear into Q/K/V:  W' = W @ Win (32x3),
//           b' = W @ bin + b.  384 outputs total -> d_ws.
// Layout of cw: [Wq' 96][Wk' 96][Wv' 96][bq' 32][bk' 32][bv' 32]
// ---------------------------------------------------------------------------
__global__ __launch_bounds__(384) void compose_weights_kernel(
    const float* __restrict__ Win, const float* __restrict__ bin,
    const float* __restrict__ Wq,  const float* __restrict__ bq,
    const float* __restrict__ Wk,  const float* __restrict__ bk,
    const float* __restrict__ Wv,  const float* __restrict__ bv,
    float* __restrict__ cw)
{
  const int t = threadIdx.x;            // 0..383
  if (t < 288) {
    const int m = t / 96;               // 0=q 1=k 2=v
    const int e = t - m * 96;
    const int o = e / 3, c = e - o * 3;
    const float* W = (m == 0) ? Wq : (m == 1) ? Wk : Wv;
    float acc = 0.f;
#pragma unroll
    for (int i = 0; i < DM; ++i) acc += W[o * DM + i] * Win[i * CC + c];
    cw[t] = acc;
  } else {
    const int e = t - 288;
    const int m = e / 32;
    const int o = e - m * 32;
    const float* W  = (m == 0) ? Wq : (m == 1) ? Wk : Wv;
    const float* bb = (m == 0) ? bq : (m == 1) ? bk : bv;
    float acc = bb[o];
#pragma unroll
    for (int i = 0; i < DM; ++i) acc += W[o * DM + i] * bin[i];
    cw[t] = acc;
  }
}

// ---------------------------------------------------------------------------
// Kernel 2: one wave = 16 positions.  Per t: K_t,V_t = z_t @ W'^T + b' in a
// single V_WMMA_F32_16X16X4_F32 per 16-channel half: K=0..2 carry the three
// z channels, the padded K=3 slot carries (A=1.0, B=bias) so C is inline 0.
// Q only at t=7.  Then per-head softmax over 8 keys + 3-channel projection.
// ---------------------------------------------------------------------------
__global__ __launch_bounds__(32) void attn_main_kernel(
    const float* __restrict__ z,  const float* __restrict__ cw,
    const float* __restrict__ Wo, const float* __restrict__ bo,
    float* __restrict__ out)
{
  __shared__ float sCW[3 * 96];          // composed W' [mat][o*3+c]
  __shared__ float sCB[3 * 32];          // composed biases [mat][o]
  __shared__ float sWo[CC * DM];
  __shared__ float sBo[CC];
  __shared__ float sZ[TT * CC * POSW];   // [t*3+c][pos]
  __shared__ float sK[TT * POSW * DM];   // [t][pos][ch]
  __shared__ float sV[TT * POSW * DM];
  __shared__ float sQ[POSW * DM];        // t=7 query
  __shared__ float sO[POSW * DM];        // attention output

  const int l  = threadIdx.x;
  const int ln = l & 15;
  const int hi = l >> 4;

  // ---- weights to LDS ----
  for (int i = l; i < 288; i += 32) sCW[i] = cw[i];
  for (int i = l; i < 96;  i += 32) { sCB[i] = cw[288 + i]; sWo[i] = Wo[i]; }
  if (l < CC) sBo[l] = bo[l];

  // ---- this wave's 16 positions (never straddles a batch: 65536 % 16 == 0) ----
  const int posg0 = blockIdx.x * POSW;
  const int b  = posg0 >> 16;
  const int hw = posg0 & (HWN - 1);

  // ---- z tile: 16 pos x (8t x 3c) = 384 floats ----
#pragma unroll
  for (int i = 0; i < 12; ++i) {
    int e  = i * 32 + l;
    int p  = e & 15;
    int tc = e >> 4;                     // 0..23
    int t  = tc / 3, c = tc - t * 3;
    sZ[tc * POSW + p] = z[(size_t)((b * TT + t) * CC + c) * HWN + hw + p];
  }
  __syncthreads();

  // ---- B fragments (4x16 f32): VGPR0 K=2*hi, VGPR1 K=2*hi+1.
  //      K=3 row (hi lanes, VGPR1) carries the composed bias. ----
  auto loadB = [&](int mat, int half) -> v2f {
    const int   n  = half * 16 + ln;
    const float* Wp = sCW + mat * 96 + n * 3;
    float w1   = Wp[1];                  // unconditional load
    float bias = sCB[mat * 32 + n];
    v2f r;
    r[0] = Wp[2 * hi];                   // K=0 -> c0  |  K=2 -> c2
    r[1] = hi ? bias : w1;               // K=3 -> b'  |  K=1 -> c1
    return r;
  };
  const v2f fq0 = loadB(0, 0), fq1 = loadB(0, 1);
  const v2f fk0 = loadB(1, 0), fk1 = loadB(1, 1);
  const v2f fv0 = loadB(2, 0), fv1 = loadB(2, 1);

  // ---- per-t: A fragment (16x4 f32) straight from z; 4 WMMAs (+2 at t=7) ----
  for (int t = 0; t < TT; ++t) {
    float z1 = sZ[(t * 3 + 1) * POSW + ln];         // unconditional load
    v2f a;
    a[0] = sZ[(t * 3 + 2 * hi) * POSW + ln];        // K=0 -> c0 | K=2 -> c2
    a[1] = hi ? 1.0f : z1;                          // K=3 -> 1  | K=1 -> c1

    v8f k0 = wmma_f32k4(a, fk0);
    v8f k1 = wmma_f32k4(a, fk1);
    v8f v0 = wmma_f32k4(a, fv0);
    v8f v1 = wmma_f32k4(a, fv1);

    // D layout: VGPR r -> pos = r + 8*hi ; N = ln (+16 for second half)
#pragma unroll
    for (int r = 0; r < 8; ++r) {
      int p = r + 8 * hi;
      sK[(t * POSW + p) * DM + ln]      = k0[r];
      sK[(t * POSW + p) * DM + 16 + ln] = k1[r];
      sV[(t * POSW + p) * DM + ln]      = v0[r];
      sV[(t * POSW + p) * DM + 16 + ln] = v1[r];
    }
    if (t == TT - 1) {
      v8f q0 = wmma_f32k4(a, fq0);
      v8f q1 = wmma_f32k4(a, fq1);
#pragma unroll
      for (int r = 0; r < 8; ++r) {
        int p = r + 8 * hi;
        sQ[p * DM + ln]      = q0[r];
        sQ[p * DM + 16 + ln] = q1[r];
      }
    }
  }
  __builtin_amdgcn_wave_barrier();

  // ---- attention for the t=7 query: 16 pos x 4 heads = 64 tasks, 2/lane ----
  const float scale = 0.35355339059327373f;         // 1/sqrt(8)
#pragma unroll
  for (int task = 0; task < 2; ++task) {
    int id = l + task * 32;
    int p  = id >> 2;
    int h  = id & 3;
    float q[8];
#pragma unroll
    for (int d = 0; d < 8; ++d) q[d] = sQ[p * DM + h * 8 + d];
    float sc[8];
    float m = -3.0e38f;
#pragma unroll
    for (int s = 0; s < 8; ++s) {
      float dot = 0.f;
#pragma unroll
      for (int d = 0; d < 8; ++d)
        dot += q[d] * sK[(s * POSW + p) * DM + h * 8 + d];
      sc[s] = dot * scale;
      m = fmaxf(m, sc[s]);
    }
    float wsum = 0.f, w[8];
#pragma unroll
    for (int s = 0; s < 8; ++s) { w[s] = __expf(sc[s] - m); wsum += w[s]; }
    const float inv = 1.f / wsum;
    float o[8];
#pragma unroll
    for (int d = 0; d < 8; ++d) o[d] = 0.f;
#pragma unroll
    for (int s = 0; s < 8; ++s) {
      float ws = w[s] * inv;
#pragma unroll
      for (int d = 0; d < 8; ++d)
        o[d] += ws * sV[(s * POSW + p) * DM + h * 8 + d];
    }
#pragma unroll
    for (int d = 0; d < 8; ++d) sO[p * DM + h * 8 + d] = o[d];
  }
  __builtin_amdgcn_wave_barrier();

  // ---- output projection: 16 pos x 3 outs = 48 tasks ----
#pragma unroll
  for (int task = 0; task < 2; ++task) {
    int e = l + task * 32;
    if (e < 48) {
      int p = e / 3, o = e - p * 3;
      float acc = sBo[o];
#pragma unroll
      for (int i = 0; i < DM; ++i) acc += sWo[o * DM + i] * sO[p * DM + i];
      out[((size_t)(b * CC + o) << 16) + hw + p] = acc;   // (B,3,H,W)
    }
  }
}

extern "C" void kernel_launch(void* const* d_in, const int* in_sizes, int n_in,
                              void* d_out, int out_size, void* d_ws, size_t ws_size,
                              hipStream_t stream) {
  (void)in_sizes; (void)n_in; (void)ws_size; (void)out_size;
  const float* z   = (const float*)d_in[0];
  const float* Win = (const float*)d_in[1];
  const float* bin = (const float*)d_in[2];
  const float* Wq  = (const float*)d_in[3];
  const float* bq  = (const float*)d_in[4];
  const float* Wk  = (const float*)d_in[5];
  const float* bk  = (const float*)d_in[6];
  const float* Wv  = (const float*)d_in[7];
  const float* bv  = (const float*)d_in[8];
  const float* Wo  = (const float*)d_in[9];
  const float* bo  = (const float*)d_in[10];
  float* outp = (float*)d_out;
  float* cw   = (float*)d_ws;                      // 384 floats

  compose_weights_kernel<<<1, 384, 0, stream>>>(Win, bin, Wq, bq, Wk, bk, Wv, bv, cw);

  const int total_pos = 4 * HWN;                   // 262144
  attn_main_kernel<<<total_pos / POSW, 32, 0, stream>>>(z, cw, Wo, bo, outp);
}